// DWT_37864431682435
// MI455X (gfx1250) — compile-verified
//
#include <hip/hip_runtime.h>
#include <cstdint>

// Toggle for the CDNA5 async global->LDS staging path.
#define DWT_ASYNC_LDS 1

typedef float v4f __attribute__((ext_vector_type(4)));

// Problem shape (fixed by the reference).
constexpr int BATCH = 8;
constexpr int CH    = 64;
constexpr int H     = 512;
constexpr int W     = 512;
constexpr int HO    = H / 2;   // 256
constexpr int WO    = W / 2;   // 256
constexpr long SUBBAND_STRIDE = (long)CH * HO * WO;       // 4,194,304 elements
constexpr long IN_ELEMS  = (long)BATCH * CH * H * W;      // 134,217,728
constexpr long N_THREADS = IN_ELEMS / 16;                 // 16 input floats / thread

__global__ __launch_bounds__(256) void dwt_haar_kernel(const float* __restrict__ x,
                                                       float* __restrict__ out) {
    const unsigned t  = blockIdx.x * 256u + threadIdx.x;
    const unsigned wq = t & 63u;           // output col quad index (wo = wq*4)
    const unsigned ho = (t >> 6) & 255u;   // output row
    const unsigned c  = (t >> 14) & 63u;   // channel
    const unsigned b  = t >> 20;           // batch

    // Input: two adjacent rows, 8 consecutive floats each (32 B, 16B-aligned).
    const long in_off = ((long)(b * CH + c) * H + 2 * ho) * (long)W + (wq << 3);
    const float* r0 = x + in_off;      // even row
    const float* r1 = r0 + W;          // odd row

    float4 a4, b4, c4, d4;

#if DWT_ASYNC_LDS && defined(__HIP_DEVICE_COMPILE__) && defined(__gfx1250__)
    // Stage 64 B/thread through LDS with the CDNA5 async data mover.
    // INST_OFFSET applies to BOTH the LDS and global addresses, so each
    // (lds,row) pair covers its two 16B beats with offset:0 / offset:16.
    // Input is read exactly once -> non-temporal to avoid thrashing the 192MB L2.
    __shared__ float tile[256 * 16];   // 16 KB per block
    float* my = tile + threadIdx.x * 16;
    uint32_t l0 = (uint32_t)(uintptr_t)my;        // LDS byte address (low 32 bits)
    uint32_t l1 = l0 + 32u;
    asm volatile(
        "global_load_async_to_lds_b128 %0, %2, off th:TH_LOAD_NT\n\t"
        "global_load_async_to_lds_b128 %0, %2, off offset:16 th:TH_LOAD_NT\n\t"
        "global_load_async_to_lds_b128 %1, %3, off th:TH_LOAD_NT\n\t"
        "global_load_async_to_lds_b128 %1, %3, off offset:16 th:TH_LOAD_NT\n\t"
        "s_wait_asynccnt 0x0"
        :
        : "v"(l0), "v"(l1), "v"(r0), "v"(r1)
        : "memory");
    const float4* mv = (const float4*)my;
    a4 = mv[0];  // row0 cols 0..3
    b4 = mv[1];  // row0 cols 4..7
    c4 = mv[2];  // row1 cols 0..3
    d4 = mv[3];  // row1 cols 4..7
#else
    a4 = ((const float4*)r0)[0];
    b4 = ((const float4*)r0)[1];
    c4 = ((const float4*)r1)[0];
    d4 = ((const float4*)r1)[1];
#endif

    const float e[8] = {a4.x, a4.y, a4.z, a4.w, b4.x, b4.y, b4.z, b4.w}; // even row
    const float f[8] = {c4.x, c4.y, c4.z, c4.w, d4.x, d4.y, d4.z, d4.w}; // odd row

    float LL[4], LH[4], HL[4], HH[4];
#pragma unroll
    for (int p = 0; p < 4; ++p) {
        // x1 = even/even, x2 = odd/even, x3 = even/odd, x4 = odd/odd
        const float x1 = e[2 * p],     x3 = e[2 * p + 1];
        const float x2 = f[2 * p],     x4 = f[2 * p + 1];
        const float s14 = x1 + x4, s23 = x2 + x3;
        LL[p] = 0.5f * (s14 + s23);
        HH[p] = 0.5f * (s14 - s23);
        LH[p] = 0.5f * ((x2 + x4) - (x1 + x3));
        HL[p] = 0.5f * ((x3 + x4) - (x1 + x2));
    }

    // Output NCHW [8, 256, 256, 256]; subband s lives at channel s*64 + c.
    const long obase = ((long)(b * 4 * CH + c) * HO + ho) * (long)WO + (wq << 2);
    float* o0 = out + obase;
    float* o1 = o0 + SUBBAND_STRIDE;
    float* o2 = o0 + 2 * SUBBAND_STRIDE;
    float* o3 = o0 + 3 * SUBBAND_STRIDE;

    v4f ll = {LL[0], LL[1], LL[2], LL[3]};
    v4f lh = {LH[0], LH[1], LH[2], LH[3]};
    v4f hl = {HL[0], HL[1], HL[2], HL[3]};
    v4f hh = {HH[0], HH[1], HH[2], HH[3]};

#if defined(__HIP_DEVICE_COMPILE__) && defined(__gfx1250__)
    // Write-once streaming output -> non-temporal stores (single b128 each).
    asm volatile("global_store_b128 %0, %1, off th:TH_STORE_NT" :: "v"(o0), "v"(ll) : "memory");
    asm volatile("global_store_b128 %0, %1, off th:TH_STORE_NT" :: "v"(o1), "v"(lh) : "memory");
    asm volatile("global_store_b128 %0, %1, off th:TH_STORE_NT" :: "v"(o2), "v"(hl) : "memory");
    asm volatile("global_store_b128 %0, %1, off th:TH_STORE_NT" :: "v"(o3), "v"(hh) : "memory");
#else
    *(v4f*)o0 = ll;
    *(v4f*)o1 = lh;
    *(v4f*)o2 = hl;
    *(v4f*)o3 = hh;
#endif
}

extern "C" void kernel_launch(void* const* d_in, const int* in_sizes, int n_in,
                              void* d_out, int out_size, void* d_ws, size_t ws_size,
                              hipStream_t stream) {
    (void)in_sizes; (void)n_in; (void)out_size; (void)d_ws; (void)ws_size;
    const float* x = (const float*)d_in[0];
    float* out = (float*)d_out;
    const int block = 256;
    const long grid = N_THREADS / block;   // 32768 blocks
    dwt_haar_kernel<<<dim3((unsigned)grid), dim3(block), 0, stream>>>(x, out);
}